// ModelNew_66924180406467
// MI455X (gfx1250) — compile-verified
//
#include <hip/hip_runtime.h>
#include <math.h>

typedef float v2f __attribute__((ext_vector_type(2)));
typedef float v8f __attribute__((ext_vector_type(8)));

#define IN_DIM 4096

// ---------------------------------------------------------------------------
// Kernel 1: wbar[i] = (1/OUT) * sum_o W[o][i] computed on the matrix pipe with
// V_WMMA_F32_16X16X4_F32 (A = all-ones).  Also wbar_c[IN_DIM] = mean(b - sub).
// Blocks 0..63: 4 waves x 16 columns each -> 4096 columns. Block 64: bias mean.
// ---------------------------------------------------------------------------
__global__ void __launch_bounds__(128)
prep_wbar_kernel(const float* __restrict__ W, const float* __restrict__ bias,
                 const float* __restrict__ sub, float* __restrict__ wbar_c,
                 int OUT) {
  __shared__ float sred[4];
  if (blockIdx.x == (IN_DIM / 16) / 4) {
    // ---- bias/subtract mean (block-uniform branch) ----
    float p = 0.f;
    for (int i = threadIdx.x; i < OUT; i += 128) p += bias[i] - sub[i];
#pragma unroll
    for (int off = 16; off > 0; off >>= 1) p += __shfl_xor(p, off, 32);
    const int lane = threadIdx.x & 31, w = threadIdx.x >> 5;
    if (lane == 0) sred[w] = p;
    __syncthreads();
    if (threadIdx.x == 0)
      wbar_c[IN_DIM] = (sred[0] + sred[1] + sred[2] + sred[3]) / (float)OUT;
    return;
  }

  const int wave = threadIdx.x >> 5;      // 4 waves per block
  const int lane = threadIdx.x & 31;      // wave32
  const int i0   = (blockIdx.x * 4 + wave) * 16;   // this wave's 16 columns
  const int col  = i0 + (lane & 15);
  const int half = lane >> 4;             // lanes 16-31 hold the other K pair

  v8f acc = {0.f, 0.f, 0.f, 0.f, 0.f, 0.f, 0.f, 0.f};
  v2f a;  a[0] = 1.0f;  a[1] = 1.0f;      // A = ones(16x4)

  for (int k0 = 0; k0 < OUT; k0 += 4) {
    const size_t kb = (size_t)(k0 + 2 * half);
    v2f b;
    b[0] = W[kb       * IN_DIM + col];    // B = 4x16 tile of W (column = lane%16)
    b[1] = W[(kb + 1) * IN_DIM + col];
    // D = A*B + C : every row of D holds the column sums
    acc = __builtin_amdgcn_wmma_f32_16x16x4_f32(
        /*neg_a=*/false, a, /*neg_b=*/false, b,
        /*c_mod=*/(short)0, acc, /*reuse_a=*/false, /*reuse_b=*/false);
  }
  // C/D layout: VGPR0, lanes 0-15 = D[M=0][N=lane]
  if (lane < 16) wbar_c[i0 + lane] = acc[0] / (float)OUT;
}

// ---------------------------------------------------------------------------
// Kernel 2 (bandwidth-bound streaming pass, ~268 MB HBM):
//   one block per row: m = x[row].wbar + c ; g = gelu_erf(m) ; out = x + g
// ---------------------------------------------------------------------------
__global__ void __launch_bounds__(256)
fused_row_kernel(const float* __restrict__ x, const float* __restrict__ wbar_c,
                 float* __restrict__ out) {
  const size_t row  = blockIdx.x;
  const float* xr   = x   + row * IN_DIM;
  float*       orow = out + row * IN_DIM;
  const int t = threadIdx.x;

  float4 xv[4];
  float partial = 0.f;
#pragma unroll
  for (int j = 0; j < 4; ++j) {           // perfectly coalesced float4 stream
    const int e = (j * 256 + t) * 4;
    xv[j] = *reinterpret_cast<const float4*>(xr + e);
    const float4 wv = *reinterpret_cast<const float4*>(wbar_c + e);
    partial += xv[j].x * wv.x + xv[j].y * wv.y + xv[j].z * wv.z + xv[j].w * wv.w;
  }

  // wave32 reduce, then across the 8 waves via LDS
#pragma unroll
  for (int off = 16; off > 0; off >>= 1) partial += __shfl_xor(partial, off, 32);
  __shared__ float sred[8];
  const int lane = t & 31, w = t >> 5;
  if (lane == 0) sred[w] = partial;
  __syncthreads();

  float m = wbar_c[IN_DIM];               // + mean(b - subtract)
#pragma unroll
  for (int i = 0; i < 8; ++i) m += sred[i];

  // exact erf-based GELU (matches jax.nn.gelu(approximate=False))
  const float g = 0.5f * m * (1.0f + erff(m * 0.70710678118654752440f));

#pragma unroll
  for (int j = 0; j < 4; ++j) {
    const int e = (j * 256 + t) * 4;
    float4 o = xv[j];
    o.x += g; o.y += g; o.z += g; o.w += g;
    *reinterpret_cast<float4*>(orow + e) = o;
  }
}

// ---------------------------------------------------------------------------
extern "C" void kernel_launch(void* const* d_in, const int* in_sizes, int n_in,
                              void* d_out, int out_size, void* d_ws, size_t ws_size,
                              hipStream_t stream) {
  const float* x   = (const float*)d_in[0];   // (B, 4096)
  const float* W   = (const float*)d_in[1];   // (OUT, 4096) row-major
  const float* b   = (const float*)d_in[2];   // (OUT,)
  const float* sub = (const float*)d_in[3];   // (OUT,)
  float* out    = (float*)d_out;
  float* wbar_c = (float*)d_ws;               // IN_DIM + 1 floats of scratch

  const int OUT = in_sizes[2];                // 1024
  const int B   = in_sizes[0] / IN_DIM;       // 8192

  prep_wbar_kernel<<<(IN_DIM / 16) / 4 + 1, 128, 0, stream>>>(W, b, sub, wbar_c, OUT);
  fused_row_kernel<<<B, 256, 0, stream>>>(x, wbar_c, out);
}